// Combine_Attention_Layer_16484084483854
// MI455X (gfx1250) — compile-verified
//
#include <hip/hip_runtime.h>
#include <math.h>

// Problem constants (from reference): B=8, N=512, NW=48, H=8, D=6
#define B_ 8
#define N_ 512
#define NW_ 48
#define H_ 8
#define D_ 6

typedef float v2f __attribute__((ext_vector_type(2)));
typedef float v8f __attribute__((ext_vector_type(8)));

// D = A(16x4) * B(4x16) + C(16x16), all f32, wave32
#define WMMA_F32_4(a, b, c) \
    __builtin_amdgcn_wmma_f32_16x16x4_f32(false, (a), false, (b), (short)0, (c), false, false)

__device__ __forceinline__ float sigmoidf(float x) {
    return 1.0f / (1.0f + __expf(-x));
}

// Generic pointer into LDS -> wave-relative LDS byte address (low 32 bits of
// the flat address; the LDS aperture base occupies addr[63:32] only).
__device__ __forceinline__ uint32_t lds_addr(const void* p) {
    return (uint32_t)(uintptr_t)p;
}

// CDNA5 async Global->LDS copy, 16B per lane (ASYNCcnt-tracked).
__device__ __forceinline__ void async_ld128(uint32_t lds_off, const float* g) {
    asm volatile("global_load_async_to_lds_b128 %0, %1, off"
                 :: "v"(lds_off), "v"(g) : "memory");
}
// CDNA5 async LDS->Global copy, 16B per lane.
__device__ __forceinline__ void async_st128(float* g, uint32_t lds_off) {
    asm volatile("global_store_async_from_lds_b128 %0, %1, off"
                 :: "v"(g), "v"(lds_off) : "memory");
}
__device__ __forceinline__ void wait_async0() {
    asm volatile("s_wait_asynccnt 0x0" ::: "memory");
}

// ---------------------------------------------------------------------------
// Kernel 1: qkv = X @ W + bias   for X in {s, t}
//   X: (B*N, NW=48), W: (48, 144), out: (B*N, 144)
//   grid = (256 row-tiles, 9 col-tiles, 2 {s,t}), block = 1 wave (32)
// ---------------------------------------------------------------------------
__global__ __launch_bounds__(32) void qkv_proj_kernel(
    const float* __restrict__ s, const float* __restrict__ t,
    const float* __restrict__ Ws, const float* __restrict__ bs,
    const float* __restrict__ Wt, const float* __restrict__ bt,
    float* __restrict__ qkvs, float* __restrict__ qkvt)
{
    const int lane = threadIdx.x;
    const int lo16 = lane & 15;
    const int hi   = lane >> 4;          // 0 or 1

    const bool isT = (blockIdx.z != 0);
    const float* __restrict__ X    = isT ? t  : s;
    const float* __restrict__ W    = isT ? Wt : Ws;
    const float* __restrict__ bias = isT ? bt : bs;
    float* __restrict__ O          = isT ? qkvt : qkvs;

    const int row0 = blockIdx.x * 16;    // over B*N = 4096
    const int col0 = blockIdx.y * 16;    // over 3*NW = 144
    const int arow = row0 + lo16;        // A-operand row for this lane
    const int bcol = col0 + lo16;        // B-operand col for this lane

    v8f acc = {};
    #pragma unroll
    for (int kk = 0; kk < 12; ++kk) {    // K = 48 in steps of 4
        const int kb = kk * 4 + hi * 2;
        v2f a, b;
        a.x = X[(long)arow * NW_ + kb];
        a.y = X[(long)arow * NW_ + kb + 1];
        b.x = W[(long)(kb)     * 144 + bcol];
        b.y = W[(long)(kb + 1) * 144 + bcol];
        acc = WMMA_F32_4(a, b, acc);
    }

    const float bv = bias[bcol];
    #pragma unroll
    for (int r = 0; r < 8; ++r) {
        const int orow = row0 + r + hi * 8;      // C/D layout: M = r + 8*hi
        O[(long)orow * 144 + bcol] = acc[r] + bv;
    }
}

// ---------------------------------------------------------------------------
// Kernel 2: fused attention.
//   Block = 256 threads = 8 waves, one wave per head. One block per (b, l-tile),
//   looping over m-tiles. All global tile traffic (E/G both orientations, mask,
//   Dst both orientations, As_hat out) moves as coalesced 16B chunks on the
//   CDNA5 async Global<->LDS path; per-head strided access happens in LDS.
//   Tile loads for iteration mt+1 are issued as soon as iteration mt's waves
//   have all finished reading the tile buffers, overlapping the async DMA with
//   the As-store + Y-GEMM work.
//   qkv layout: [(b*N + n)*144 + comp*8 + h], comp 0..5 = Q, 6..11 = K, 12..17 = V
// ---------------------------------------------------------------------------
__global__ __launch_bounds__(256) void attn_kernel(
    const float* __restrict__ Dst, const float* __restrict__ E,
    const float* __restrict__ G,   const float* __restrict__ mask_s,
    const float* __restrict__ qkvs, const float* __restrict__ qkvt,
    float* __restrict__ outY, float* __restrict__ outAs)
{
    // Tiles: [l][m][h] (lm) or [m][l][h] (ml), 16*16*8 floats = 8KB each
    __shared__ float eLM[2048], eML[2048];
    __shared__ float gLM[2048], gML[2048];
    __shared__ float mLM[2048];
    __shared__ float asLM[2048];           // As_hat staging for async store-out
    __shared__ float dLM[256], dML[256];   // Dst tiles (16x16)
    __shared__ float Ps[H_][256], Pt[H_][256];  // per-head P tiles

    const int tid  = threadIdx.x;          // 0..255
    const int h    = tid >> 5;             // wave id == head
    const int lane = tid & 31;
    const int lo16 = lane & 15;
    const int hi   = lane >> 4;
    const int l0 = blockIdx.x * 16;
    const int b  = blockIdx.y;

    const float scale = rsqrtf((float)D_);

    // Issue the cooperative async tile loads for m-tile starting at m0.
    auto issue_tile_loads = [&](int m0) {
        #pragma unroll
        for (int q = 0; q < 2; ++q) {
            const int c   = tid + q * 256;     // chunk id 0..511
            const int row = c >> 5;            // tile row (l for LM, m for ML)
            const int col = (c & 31) * 4;      // float offset within 128-float row
            const long offLM = ((long)(b * N_ + l0 + row)) * N_ * H_ + (long)m0 * H_ + col;
            const long offML = ((long)(b * N_ + m0 + row)) * N_ * H_ + (long)l0 * H_ + col;
            async_ld128(lds_addr(&eLM[c * 4]), E + offLM);
            async_ld128(lds_addr(&eML[c * 4]), E + offML);
            async_ld128(lds_addr(&gLM[c * 4]), G + offLM);
            async_ld128(lds_addr(&gML[c * 4]), G + offML);
            async_ld128(lds_addr(&mLM[c * 4]), mask_s + offLM);
        }
        if (tid < 64) {                        // Dst tiles: 64 float4 chunks each
            const int row = tid >> 2;
            const int col = (tid & 3) * 4;
            async_ld128(lds_addr(&dLM[tid * 4]),
                        Dst + ((long)(b * N_ + l0 + row)) * N_ + m0 + col);
            async_ld128(lds_addr(&dML[tid * 4]),
                        Dst + ((long)(b * N_ + m0 + row)) * N_ + l0 + col);
        }
    };

    // ---- A operands (persist across m loop): Qs/Qt rows l0..l0+15, head h ----
    const int arow = l0 + lo16;
    const long qrow = ((long)(b * N_ + arow)) * 144 + h;   // +comp*8 selects element
    v2f aQs[2], aQt[2];
    #pragma unroll
    for (int kk = 0; kk < 2; ++kk) {
        const int d0 = kk * 4 + hi * 2;
        const int d1 = d0 + 1;
        v2f as_, at_;
        as_.x = (d0 < D_) ? qkvs[qrow + (long)d0 * H_] : 0.0f;
        as_.y = (d1 < D_) ? qkvs[qrow + (long)d1 * H_] : 0.0f;
        at_.x = (d0 < D_) ? qkvt[qrow + (long)d0 * H_] : 0.0f;
        at_.y = (d1 < D_) ? qkvt[qrow + (long)d1 * H_] : 0.0f;
        aQs[kk] = as_;
        aQt[kk] = at_;
    }

    v8f yacc = {};

    issue_tile_loads(0);       // prologue: prefetch first m-tile

    for (int mt = 0; mt < N_ / 16; ++mt) {
        const int m0 = mt * 16;

        wait_async0();        // own tile loads landed; own prev As-store drained
        __syncthreads();      // whole block's tiles ready, all As-stores drained

        // ---- Score GEMMs: cs = Qs·Kt^T, ct = Qt·Ks^T (K = D=6, padded to 8) ----
        const int brow = m0 + lo16;            // this lane's B column = m
        const long krow = ((long)(b * N_ + brow)) * 144 + h;
        v8f cs = {}, ct = {};
        #pragma unroll
        for (int kk = 0; kk < 2; ++kk) {
            const int d0 = kk * 4 + hi * 2;
            const int d1 = d0 + 1;
            v2f bKt, bKs;
            bKt.x = (d0 < D_) ? qkvt[krow + (long)(D_ + d0) * H_] : 0.0f;
            bKt.y = (d1 < D_) ? qkvt[krow + (long)(D_ + d1) * H_] : 0.0f;
            bKs.x = (d0 < D_) ? qkvs[krow + (long)(D_ + d0) * H_] : 0.0f;
            bKs.y = (d1 < D_) ? qkvs[krow + (long)(D_ + d1) * H_] : 0.0f;
            cs = WMMA_F32_4(aQs[kk], bKt, cs);
            ct = WMMA_F32_4(aQt[kk], bKs, ct);
        }

        // ---- Elementwise (all operands from LDS), build As tile + P tiles ----
        #pragma unroll
        for (int r = 0; r < 8; ++r) {
            const int lL = r + hi * 8;         // C/D layout row (local l)
            const int mL = lo16;               // C/D layout col (local m)
            const int lm = lL * 16 + mL;
            const int ml = mL * 16 + lL;

            const float as_hat = cs[r] * scale + eLM[lm * 8 + h];
            const float at_hat = ct[r] * scale + eML[ml * 8 + h];
            asLM[lm * 8 + h] = as_hat;

            const float msk = mLM[lm * 8 + h];
            const float ps  = sigmoidf(as_hat * dLM[lm] - at_hat * dML[ml]);
            const float gs  = sigmoidf(gLM[lm * 8 + h] + msk);
            const float gt  = sigmoidf(gML[ml * 8 + h] + msk);

            Ps[h][lm] = ps * gs;
            Pt[h][lm] = (1.0f - ps) * gt;
        }
        __syncthreads();      // tiles fully consumed + asLM fully built

        // ---- Async store As_hat tile: contiguous 512B rows ----
        #pragma unroll
        for (int q = 0; q < 2; ++q) {
            const int c   = tid + q * 256;
            const int row = c >> 5;
            const int col = (c & 31) * 4;
            float* g = outAs + (((long)(b * N_ + l0 + row)) * N_ + m0) * H_ + col;
            async_st128(g, lds_addr(&asLM[c * 4]));
        }

        // ---- Prefetch next m-tile: overlaps with As-store + Y GEMMs ----
        if (mt + 1 < N_ / 16) issue_tile_loads(m0 + 16);

        // ---- Y GEMMs: yacc += P_s @ Vt + P_t @ Vs (K = 16 over m, steps of 4) ----
        #pragma unroll
        for (int kk = 0; kk < 4; ++kk) {
            const int kb = kk * 4 + hi * 2;
            v2f aPs, aPt, bVt, bVs;
            aPs.x = Ps[h][lo16 * 16 + kb];
            aPs.y = Ps[h][lo16 * 16 + kb + 1];
            aPt.x = Pt[h][lo16 * 16 + kb];
            aPt.y = Pt[h][lo16 * 16 + kb + 1];

            const bool nv = (lo16 < D_);       // output col = V's d index
            const long vb0 = ((long)(b * N_ + m0 + kb))     * 144 + h;
            const long vb1 = ((long)(b * N_ + m0 + kb + 1)) * 144 + h;
            bVt.x = nv ? qkvt[vb0 + (long)(2 * D_ + lo16) * H_] : 0.0f;
            bVt.y = nv ? qkvt[vb1 + (long)(2 * D_ + lo16) * H_] : 0.0f;
            bVs.x = nv ? qkvs[vb0 + (long)(2 * D_ + lo16) * H_] : 0.0f;
            bVs.y = nv ? qkvs[vb1 + (long)(2 * D_ + lo16) * H_] : 0.0f;

            yacc = WMMA_F32_4(aPs, bVt, yacc);
            yacc = WMMA_F32_4(aPt, bVs, yacc);
        }
        // next iteration's wait_async0 + barrier guards tile/asLM reuse
    }

    wait_async0();            // drain final As_hat async store

    // ---- Store Y: (b, l, h*D + k), k = lo16 < 6 ----
    if (lo16 < D_) {
        #pragma unroll
        for (int r = 0; r < 8; ++r) {
            const int l = l0 + r + hi * 8;
            outY[((long)(b * N_ + l)) * (H_ * D_) + h * D_ + lo16] = yacc[r];
        }
    }
}

// ---------------------------------------------------------------------------
// Launch
// ---------------------------------------------------------------------------
extern "C" void kernel_launch(void* const* d_in, const int* in_sizes, int n_in,
                              void* d_out, int out_size, void* d_ws, size_t ws_size,
                              hipStream_t stream) {
    (void)in_sizes; (void)n_in; (void)out_size; (void)ws_size;

    const float* Dst   = (const float*)d_in[1];
    const float* s     = (const float*)d_in[2];
    const float* t     = (const float*)d_in[3];
    const float* E     = (const float*)d_in[4];
    const float* G     = (const float*)d_in[5];
    const float* mask_ = (const float*)d_in[6];
    const float* Ws    = (const float*)d_in[7];
    const float* bs    = (const float*)d_in[8];
    const float* Wt    = (const float*)d_in[9];
    const float* bt    = (const float*)d_in[10];

    float* outY  = (float*)d_out;                       // B*N*48
    float* outAs = outY + (size_t)B_ * N_ * (H_ * D_);  // B*N*N*H

    float* qkvs = (float*)d_ws;                         // B*N*144 floats
    float* qkvt = qkvs + (size_t)B_ * N_ * 144;

    qkv_proj_kernel<<<dim3((B_ * N_) / 16, 144 / 16, 2), 32, 0, stream>>>(
        s, t, Ws, bs, Wt, bt, qkvs, qkvt);

    attn_kernel<<<dim3(N_ / 16, B_), 256, 0, stream>>>(
        Dst, E, G, mask_, qkvs, qkvt, outY, outAs);
}